// RepLoss_10926396801199
// MI455X (gfx1250) — compile-verified
//
#include <hip/hip_runtime.h>
#include <math.h>

#define SIGMA      0.5f
#define EPS_F      1e-7f
#define GT_CHUNK   512
#define TILE       128
#define BLK        256
#define BOXF       5          // packed floats per box: x0,y0,x1,y1,area(+1)

#if defined(__has_builtin)
#  if __has_builtin(__builtin_amdgcn_tensor_load_to_lds)
#    define HAVE_TDM 1
#  endif
#endif
#ifndef HAVE_TDM
#  define HAVE_TDM 0
#endif

typedef unsigned int tdm_u32x4 __attribute__((ext_vector_type(4)));
typedef int          tdm_i32x4 __attribute__((ext_vector_type(4)));
typedef int          tdm_i32x8 __attribute__((ext_vector_type(8)));

// Low 32 bits of a generic LDS pointer are the LDS byte address (ISA 10.2).
__device__ __forceinline__ unsigned lds_addr_of(const void* p) {
    return (unsigned)(uintptr_t)p;
}

#if HAVE_TDM
// 1D contiguous tile load: global -> LDS via Tensor Data Mover.
// tensor_elems = elements remaining in the array from tile start (OOB -> 0),
// tile_elems   = elements to move. data_size = 4 bytes.
__device__ __forceinline__ void tdm_load_1d(unsigned lds_byte_addr,
                                            const float* gsrc,
                                            unsigned tensor_elems,
                                            unsigned tile_elems) {
    unsigned long long ga = (unsigned long long)(uintptr_t)gsrc;
    tdm_u32x4 g0;
    g0.x = 1u;                                    // count=1 (valid descriptor)
    g0.y = lds_byte_addr;                         // D#.lds_addr
    g0.z = (unsigned)(ga & 0xffffffffu);          // global_addr[31:0]
    g0.w = (unsigned)((ga >> 32) & 0x01ffffffu)   // global_addr[56:32]
         | 0x80000000u;                           // type=2 ("image")
    tdm_i32x8 g1;
    g1[0] = 0x20000;                               // data_size=2 (4B); no flags
    g1[1] = (int)((tensor_elems & 0xffffu) << 16); // tensor_dim0[15:0]
    g1[2] = (int)((tensor_elems >> 16) & 0xffffu); // tensor_dim0[31:16]; dim1=0
    g1[3] = (int)(tile_elems << 16);               // tile_dim0 (dim1 hi = 0)
    g1[4] = 0;                                     // tile_dim1=0, tile_dim2=0
    g1[5] = (int)tensor_elems;                     // tensor_dim0_stride[31:0]
    g1[6] = 0;                                     // stride0 hi, stride1 lo
    g1[7] = 0;                                     // stride1 hi
    tdm_i32x4 z4; z4[0] = 0; z4[1] = 0; z4[2] = 0; z4[3] = 0;
#if defined(__clang_major__) && (__clang_major__ >= 23)
    tdm_i32x8 z8;
    z8[0]=0; z8[1]=0; z8[2]=0; z8[3]=0; z8[4]=0; z8[5]=0; z8[6]=0; z8[7]=0;
    __builtin_amdgcn_tensor_load_to_lds(g0, g1, z4, z4, z8, 0);
#else
    __builtin_amdgcn_tensor_load_to_lds(g0, g1, z4, z4, 0);
#endif
}
#endif // HAVE_TDM

// 1 - SIGMA - ln(1-SIGMA) = 0.5 + ln(2)
__device__ __forceinline__ float rep_term(float v) {
    const float inv_den = 1.0f / (0.5f + 0.69314718055994530942f);
    float smln = (v - SIGMA) * inv_den;
    float nl   = -logf(fmaxf(1.0f - v, EPS_F));
    return (v > SIGMA) ? smln : nl;
}

__device__ __forceinline__ float smooth_l1(float a, float b) {
    float d = fabsf(a - b);
    return (d < 1.0f) ? (0.5f * d * d) : (d - 0.5f);
}

// xywh -> packed {x0,y0,x1,y1,area(+1)}; zero accumulators.
__global__ void repl_init_kernel(const float* __restrict__ gt,
                                 const float* __restrict__ pr,
                                 float* __restrict__ gpack,
                                 float* __restrict__ ppack,
                                 double* __restrict__ acc, int M, int N) {
    int idx = blockIdx.x * blockDim.x + threadIdx.x;
    if (idx < 4) acc[idx] = 0.0;
    if (idx < M) {
        __builtin_prefetch(gt + (size_t)idx * 4, 0, 0);
        float x = gt[idx*4+0], y = gt[idx*4+1], w = gt[idx*4+2], h = gt[idx*4+3];
        float x0 = x - w * 0.5f, y0 = y - h * 0.5f;
        float x1 = x + w * 0.5f, y1 = y + h * 0.5f;
        float* o = gpack + (size_t)idx * BOXF;
        o[0] = x0; o[1] = y0; o[2] = x1; o[3] = y1;
        o[4] = (x1 - x0 + 1.0f) * (y1 - y0 + 1.0f);
    }
    if (idx < N) {
        __builtin_prefetch(pr + (size_t)idx * 4, 0, 0);
        float x = pr[idx*4+0], y = pr[idx*4+1], w = pr[idx*4+2], h = pr[idx*4+3];
        float x0 = x - w * 0.5f, y0 = y - h * 0.5f;
        float x1 = x + w * 0.5f, y1 = y + h * 0.5f;
        float* o = ppack + (size_t)idx * BOXF;
        o[0] = x0; o[1] = y0; o[2] = x1; o[3] = y1;
        o[4] = (x1 - x0 + 1.0f) * (y1 - y0 + 1.0f);
    }
}

// Per-pred top-2 over GT IoUs -> Attr (SmoothL1 vs top-1) + RepGT (IoG vs top-2).
__global__ void repl_top2_kernel(const float* __restrict__ gpack,
                                 const float* __restrict__ ppack,
                                 double* __restrict__ acc, int M, int N) {
    __shared__ __align__(16) float sg[GT_CHUNK * BOXF];
    __shared__ double redA[BLK];
    __shared__ double redR[BLK];

    int t = threadIdx.x;
    int j = blockIdx.x * blockDim.x + t;
    bool active = (j < N);

    float px0 = 0.f, py0 = 0.f, px1 = 0.f, py1 = 0.f, aP = 1.f;
    if (active) {
        const float* pb = ppack + (size_t)j * BOXF;
        px0 = pb[0]; py0 = pb[1]; px1 = pb[2]; py1 = pb[3]; aP = pb[4];
    }

    float m1 = -1.0f, m2 = -1.0f;
    int   i1 = 0,     i2 = 0;

    for (int base = 0; base < M; base += GT_CHUNK) {
        int cnt = M - base; if (cnt > GT_CHUNK) cnt = GT_CHUNK;
        __syncthreads();
#if HAVE_TDM
        if (t < 32) {   // wave 0 issues the DMA for the whole block
            tdm_load_1d(lds_addr_of(sg), gpack + (size_t)base * BOXF,
                        (unsigned)((M - base) * BOXF), (unsigned)(cnt * BOXF));
            __builtin_amdgcn_s_wait_tensorcnt(0);
        }
#else
        for (int k = t; k < cnt * BOXF; k += blockDim.x)
            sg[k] = gpack[(size_t)base * BOXF + k];
#endif
        __syncthreads();
        if (active) {
            for (int i = 0; i < cnt; ++i) {
                const float* g = sg + i * BOXF;
                float w = fminf(g[2], px1) - fmaxf(g[0], px0) + 1.0f;
                float h = fminf(g[3], py1) - fmaxf(g[1], py0) + 1.0f;
                w = fmaxf(w, 0.0f); h = fmaxf(h, 0.0f);
                float ov  = w * h;
                float iou = ov / fmaxf(g[4] + aP - ov, EPS_F);
                iou = fminf(fmaxf(iou, EPS_F), 1.0f);
                int gi = base + i;
                if (iou > m1)      { m2 = m1; i2 = i1; m1 = iou; i1 = gi; }
                else if (iou > m2) { m2 = iou; i2 = gi; }
            }
        }
    }

    float attr_l = 0.0f, rep_l = 0.0f;
    if (active) {
        const float* ga = gpack + (size_t)i1 * BOXF;   // top-1 GT
        attr_l = smooth_l1(px0, ga[0]) + smooth_l1(py0, ga[1]) +
                 smooth_l1(px1, ga[2]) + smooth_l1(py1, ga[3]);
        const float* gr = gpack + (size_t)i2 * BOXF;   // top-2 GT
        float w = fminf(px1, gr[2]) - fmaxf(px0, gr[0]);   // no +1 in IoG
        float h = fminf(py1, gr[3]) - fmaxf(py0, gr[1]);
        w = fmaxf(w, 0.0f); h = fmaxf(h, 0.0f);
        float inter  = w * h;
        float g_area = fabsf(gr[2] - gr[0]) * fabsf(gr[3] - gr[1]);
        float iog    = inter / g_area;
        rep_l = rep_term(iog);
    }

    redA[t] = (double)attr_l;
    redR[t] = (double)rep_l;
    __syncthreads();
    for (int s = BLK / 2; s > 0; s >>= 1) {
        if (t < s) { redA[t] += redA[t + s]; redR[t] += redR[t + s]; }
        __syncthreads();
    }
    if (t == 0) {
        atomicAdd(&acc[0], redA[0]);
        atomicAdd(&acc[1], redR[0]);
    }
}

// RepBox: strict upper triangle of NxN pred-pred IoU. 128x128 tiles staged
// into LDS by TDM; 2 threads per tile row, 64 columns each.
__global__ void repl_repbox_kernel(const float* __restrict__ ppack,
                                   double* __restrict__ acc, int N) {
    int ti = blockIdx.y, tj = blockIdx.x;
    if (tj < ti) return;  // lower-triangle tiles contribute nothing

    __shared__ __align__(16) float bi[TILE * BOXF];
    __shared__ __align__(16) float bj[TILE * BOXF];
    __shared__ double red[BLK];

    int i0 = ti * TILE, j0 = tj * TILE;
    int t = threadIdx.x;

#if HAVE_TDM
    if (t < 32) {   // wave 0 issues both tile DMAs
        tdm_load_1d(lds_addr_of(bi), ppack + (size_t)i0 * BOXF,
                    (unsigned)((N - i0) * BOXF), (unsigned)(TILE * BOXF));
        tdm_load_1d(lds_addr_of(bj), ppack + (size_t)j0 * BOXF,
                    (unsigned)((N - j0) * BOXF), (unsigned)(TILE * BOXF));
        __builtin_amdgcn_s_wait_tensorcnt(0);
    }
#else
    for (int k = t; k < TILE * BOXF; k += blockDim.x) {
        int r = k / BOXF, c = k % BOXF;
        bi[k] = (i0 + r < N) ? ppack[(size_t)(i0 + r) * BOXF + c] : 0.0f;
        bj[k] = (j0 + r < N) ? ppack[(size_t)(j0 + r) * BOXF + c] : 0.0f;
    }
#endif
    __syncthreads();

    int r  = t >> 1;        // tile row handled by this thread
    int ch = t & 1;         // which half of the 128 columns
    float sum = 0.0f;
    int gi = i0 + r;
    if (gi < N) {
        const float* pi = bi + r * BOXF;
        float x0 = pi[0], y0 = pi[1], x1 = pi[2], y1 = pi[3], a1 = pi[4];
        int cbeg = ch * (TILE / 2), cend = cbeg + (TILE / 2);
        for (int c = cbeg; c < cend; ++c) {
            int gj = j0 + c;
            if (gj <= gi || gj >= N) continue;  // strict upper triangle
            const float* pj = bj + c * BOXF;
            float w = fminf(x1, pj[2]) - fmaxf(x0, pj[0]) + 1.0f;
            float h = fminf(y1, pj[3]) - fmaxf(y0, pj[1]) + 1.0f;
            w = fmaxf(w, 0.0f); h = fmaxf(h, 0.0f);
            float ov  = w * h;
            float iou = ov / fmaxf(a1 + pj[4] - ov, EPS_F);
            iou = fminf(fmaxf(iou, EPS_F), 1.0f);
            sum += rep_term(iou);
        }
    }

    red[t] = (double)sum;
    __syncthreads();
    for (int s = BLK / 2; s > 0; s >>= 1) {
        if (t < s) red[t] += red[t + s];
        __syncthreads();
    }
    if (t == 0) atomicAdd(&acc[2], red[0]);
}

__global__ void repl_finalize_kernel(const double* __restrict__ acc,
                                     float* __restrict__ out, int N) {
    if (blockIdx.x == 0 && threadIdx.x == 0) {
        double n     = (double)N;
        double count = n * (double)(N - 1) * 0.5;
        double v = acc[0] / n + 0.5 * (acc[1] / n) + 0.5 * (acc[2] / count);
        out[0] = (float)v;
    }
}

extern "C" void kernel_launch(void* const* d_in, const int* in_sizes, int n_in,
                              void* d_out, int out_size, void* d_ws, size_t ws_size,
                              hipStream_t stream) {
    const float* gt = (const float*)d_in[0];  // [M,4] xywh
    const float* pr = (const float*)d_in[1];  // [N,4] xywh
    int M = in_sizes[0] / 4;
    int N = in_sizes[1] / 4;

    char* ws = (char*)d_ws;
    double* acc = (double*)ws;                            // 4 doubles (32 B)
    size_t off_g = 32;
    float* gpack = (float*)(ws + off_g);                  // M*5 floats
    size_t off_p = off_g + (size_t)M * BOXF * 4;
    off_p = (off_p + 15) & ~(size_t)15;
    float* ppack = (float*)(ws + off_p);                  // N*5 floats

    int mx = (M > N) ? M : N;
    repl_init_kernel<<<(mx + BLK - 1) / BLK, BLK, 0, stream>>>(
        gt, pr, gpack, ppack, acc, M, N);

    repl_top2_kernel<<<(N + BLK - 1) / BLK, BLK, 0, stream>>>(
        gpack, ppack, acc, M, N);

    dim3 grid((N + TILE - 1) / TILE, (N + TILE - 1) / TILE);
    repl_repbox_kernel<<<grid, BLK, 0, stream>>>(ppack, acc, N);

    repl_finalize_kernel<<<1, 64, 0, stream>>>(acc, (float*)d_out, N);
}